// VectorQuantizer_34136400068857
// MI455X (gfx1250) — compile-verified
//
#include <hip/hip_runtime.h>
#include <stdint.h>

// ---------------- problem constants ----------------
#define CB_K   512
#define CDIM   64
#define PIMG   4096                  // 64*64 pixels per (b,t) image
#define NIMG   128                   // 8*16 images
#define NPIX   (NIMG * PIMG)         // 524288 pixel-vectors
#define MTILE  128                   // pixels per block
#define NBLK   (NPIX / MTILE)        // 4096 blocks
#define CBS    68                    // padded codebook row stride (floats)

// ---------------- LDS layout (float indices) ----------------
#define CB_OFF    0
#define EN_OFF    (CB_K * CBS)              // 34816
#define HIST_OFF  (EN_OFF + CB_K)           // 35328
#define TILE_OFF  (HIST_OFF + CB_K)         // 35840
#define SMEM_FLOATS (TILE_OFF + CDIM * MTILE)   // 44032
#define SMEM_BYTES  (SMEM_FLOATS * 4)           // 176128 B < 320 KB/WGP

// ---------------- output layout (floats) ----------------
#define IDX_OFF   ((size_t)NPIX * CDIM)     // 33554432
#define LOSS_OFF  (IDX_OFF + NPIX)
#define ENT_OFF   (LOSS_OFF + 1)

typedef float    v2f __attribute__((ext_vector_type(2)));
typedef float    v8f __attribute__((ext_vector_type(8)));
typedef unsigned v4u __attribute__((ext_vector_type(4)));
typedef int      v8i __attribute__((ext_vector_type(8)));
typedef int      v4i __attribute__((ext_vector_type(4)));

// ---------------- init: zero global accumulators ----------------
__global__ void vq_init(float* wsum, unsigned* hist) {
  if (threadIdx.x == 0) *wsum = 0.f;
  hist[threadIdx.x] = 0u;   // launched with 512 threads
}

// ---------------- main fused VQ kernel ----------------
__global__ __launch_bounds__(256)
void vq_main(const float* __restrict__ x, const float* __restrict__ cb,
             float* __restrict__ out, float* __restrict__ wsum,
             unsigned* __restrict__ ghist) {
  extern __shared__ float smem[];
  const int tid = threadIdx.x;
  const int blk = blockIdx.x;
  const int img = blk >> 5;            // 32 blocks per image
  const int p0  = (blk & 31) * MTILE;

  // zero enorm + local histogram (bit pattern 0 valid for both f32 and u32)
  smem[EN_OFF + tid] = 0.f;   smem[EN_OFF + 256 + tid] = 0.f;
  smem[HIST_OFF + tid] = 0.f; smem[HIST_OFF + 256 + tid] = 0.f;
  __syncthreads();

  // --- Tensor Data Mover: async 2D tile load, 128 pixels x 64 channels ---
  // Memory: channel-major rows of the image, stride 4096 floats.
  // LDS dest: planar [c][p] rows of 128 floats at TILE_OFF.
  if (tid < 32) {
    uint64_t ga = (uint64_t)(uintptr_t)(x + (size_t)img * (CDIM * PIMG) + p0);
    v4u g0;
    g0[0] = 1u;                                     // count=1 (valid descriptor)
    g0[1] = (unsigned)(TILE_OFF * 4);               // lds_addr (bytes)
    g0[2] = (unsigned)(ga & 0xFFFFFFFFull);         // global_addr[31:0]
    g0[3] = ((unsigned)(ga >> 32) & 0x01FFFFFFu) | 0x80000000u; // addr[56:32] | type=2
    v8i g1;
    g1[0] = 0x20000;                                // data_size = 4B
    g1[1] = (int)(MTILE << 16);                     // tensor_dim0 = 128
    g1[2] = (int)(CDIM  << 16);                     // tensor_dim1 = 64
    g1[3] = (int)(MTILE << 16);                     // tile_dim0   = 128
    g1[4] = CDIM;                                   // tile_dim1 = 64, tile_dim2 = 0
    g1[5] = PIMG;                                   // tensor_dim0_stride = 4096
    g1[6] = 0; g1[7] = 0;
    v4i g2 = {0, 0, 0, 0};
    v4i g3 = {0, 0, 0, 0};
    v8i g4 = {0, 0, 0, 0, 0, 0, 0, 0};
    __builtin_amdgcn_tensor_load_to_lds(g0, g1, g2, g3, g4, 0);
  }

  // --- cooperative codebook copy (overlaps TDM) + ||e_k||^2 ---
  const float4* cb4 = (const float4*)cb;
  for (int i = tid; i < CB_K * (CDIM / 4); i += 256) {
    const int row = i >> 4, q = i & 15;
    float4 v = cb4[i];
    *(float4*)&smem[CB_OFF + row * CBS + q * 4] = v;
    atomicAdd(&smem[EN_OFF + row], v.x * v.x + v.y * v.y + v.z * v.z + v.w * v.w);
  }
  if (tid < 32) __builtin_amdgcn_s_wait_tensorcnt(0);
  __syncthreads();

  // --- per-wave WMMA distance search: D^T tile = cbTile(16x64) @ pixT(64x16) ---
  const int lane = tid & 31;
  const int wv   = tid >> 5;
  const int lo   = lane & 15;
  const int hi   = lane >> 4;                 // 0 | 1
  const int pl   = wv * 16 + lo;              // local pixel 0..127
  const size_t pg = (size_t)img * PIMG + p0 + pl;  // global pixel index

  // preload B fragments (this wave's 16 pixels) once; also ||f||^2 partial
  v2f bf[16];
  float fn = 0.f;
#pragma unroll
  for (int j = 0; j < 16; ++j) {
    const int c0 = 4 * j + 2 * hi;
    float bx = smem[TILE_OFF + c0 * MTILE + pl];
    float by = smem[TILE_OFF + (c0 + 1) * MTILE + pl];
    bf[j][0] = bx; bf[j][1] = by;
    fn += bx * bx + by * by;
  }
  fn += __shfl_xor(fn, 16);                   // full ||f||^2 for pixel pl

  float bestd = 3.4e38f;
  int   bestk = 0;
  for (int kt = 0; kt < 32; ++kt) {
    const int k0 = kt * 16;
    v8f acc0 = {0, 0, 0, 0, 0, 0, 0, 0};
    v8f acc1 = {0, 0, 0, 0, 0, 0, 0, 0};
    const int abase = CB_OFF + (k0 + lo) * CBS + 2 * hi;
#pragma unroll
    for (int j = 0; j < 16; j += 2) {
      v2f a0 = *(const v2f*)&smem[abase + 4 * j];
      acc0 = __builtin_amdgcn_wmma_f32_16x16x4_f32(
          false, a0, false, bf[j], (short)0, acc0, false, false);
      v2f a1 = *(const v2f*)&smem[abase + 4 * (j + 1)];
      acc1 = __builtin_amdgcn_wmma_f32_16x16x4_f32(
          false, a1, false, bf[j + 1], (short)0, acc1, false, false);
    }
    const int kb = k0 + 8 * hi;               // row M = v + 8*hi -> entry k0+v+8*hi
#pragma unroll
    for (int v = 0; v < 8; ++v) {
      float d = smem[EN_OFF + kb + v] - 2.f * (acc0[v] + acc1[v]);
      if (d < bestd) { bestd = d; bestk = kb + v; }
    }
  }

  // combine the two half-wave candidate sets (tie-break: lowest index)
  {
    float od = __shfl_xor(bestd, 16);
    int   ok = __shfl_xor(bestk, 16);
    if (od < bestd || (od == bestd && ok < bestk)) { bestd = od; bestk = ok; }
  }
  const float err = bestd + fn;               // sum_c (e_k[c]-f[c])^2, exact

  // indices + local histogram (one writer per pixel)
  if (hi == 0) {
    out[IDX_OFF + pg] = (float)bestk;
    atomicAdd((unsigned*)(smem + HIST_OFF) + bestk, 1u);
  }

  // wave-reduce squared error, one global atomic per wave
  {
    float c = hi ? 0.f : err;
#pragma unroll
    for (int off = 16; off >= 1; off >>= 1) c += __shfl_xor(c, off);
    if (lane == 0) atomicAdd(wsum, c);
  }

  // scatter quantized vectors back in (B,T,C,H,W) layout
  const size_t qbase = (size_t)img * (CDIM * PIMG) + (size_t)(p0 + pl);
  const int cbrow = CB_OFF + bestk * CBS + 32 * hi;
#pragma unroll 8
  for (int c = 0; c < 32; ++c) {
    out[qbase + (size_t)(c + 32 * hi) * PIMG] = smem[cbrow + c];
  }

  // flush histogram
  __syncthreads();
  for (int i = tid; i < CB_K; i += 256)
    atomicAdd(&ghist[i], *((unsigned*)(smem + HIST_OFF) + i));
}

// ---------------- finalize: vq_loss + entropy ----------------
__global__ void vq_fin(const float* wsum, const unsigned* hist, float* out) {
  __shared__ float red[512];
  const unsigned c = hist[threadIdx.x];
  const float pr = (float)c / (float)NPIX;
  red[threadIdx.x] = (c > 0) ? (-pr * log2f(pr)) : 0.f;
  __syncthreads();
  for (int s = 256; s > 0; s >>= 1) {
    if ((int)threadIdx.x < s) red[threadIdx.x] += red[threadIdx.x + s];
    __syncthreads();
  }
  if (threadIdx.x == 0) {
    out[LOSS_OFF] = 1.25f * (*wsum) / (float)((size_t)NPIX * CDIM);
    out[ENT_OFF]  = red[0];
  }
}

// ---------------- host entry ----------------
extern "C" void kernel_launch(void* const* d_in, const int* in_sizes, int n_in,
                              void* d_out, int out_size, void* d_ws, size_t ws_size,
                              hipStream_t stream) {
  const float* x  = (const float*)d_in[0];   // (8,16,64,64,64) f32
  const float* cb = (const float*)d_in[1];   // (512,64) f32
  float* out  = (float*)d_out;
  float* wsum = (float*)d_ws;
  unsigned* hist = (unsigned*)((char*)d_ws + 64);

  vq_init<<<1, 512, 0, stream>>>(wsum, hist);
  vq_main<<<NBLK, 256, SMEM_BYTES, stream>>>(x, cb, out, wsum, hist);
  vq_fin<<<1, 512, 0, stream>>>(wsum, hist, out);
}